// AttentionPooledValueHead_42966852829475
// MI455X (gfx1250) — compile-verified
//
#include <hip/hip_runtime.h>
#include <cmath>

typedef __attribute__((ext_vector_type(2))) float v2f;
typedef __attribute__((ext_vector_type(4))) float v4f;
typedef __attribute__((ext_vector_type(8))) float v8f;

#define TK 64          // H-chunk staged per iteration
#define TKP (TK + 4)   // +4 floats pad -> conflict-free ds reads (row step = 4 banks mod 64)
#define WAVES 8        // waves (tiles) per block

// Pass 1: for each row s of [BS, H], compute both
//   sc[s] = dot(h_s, query) * inv_scale      (WMMA column n==0)
//   rv[s] = dot(h_s, out_w)                  (WMMA column n==1)
// One wave handles a 16-row tile with V_WMMA_F32_16X16X4_F32 accumulating over H.
__global__ __launch_bounds__(256) void attnpool_scores_kernel(
    const float* __restrict__ hs,   // [BS, H]
    const float* __restrict__ q,    // [H]
    const float* __restrict__ ow,   // [H]
    float* __restrict__ sc,         // [BS]
    float* __restrict__ rv,         // [BS]
    int BS, int H, float inv_scale)
{
  __shared__ float tile[WAVES][16 * TKP];  // 16 rows x TK floats per wave (padded)
  __shared__ float wt[WAVES][2][TKP];      // query / out_w chunks per wave

  const int w    = (int)threadIdx.x >> 5;
  const int lane = (int)threadIdx.x & 31;
  const int half = lane >> 4;   // lane-half selects K offset (A/B) and M offset (D)
  const int n    = lane & 15;   // A: row m; B/D: column n

  const int ntiles = BS >> 4;
  int tileIdx = (int)blockIdx.x * WAVES + w;
  const bool active = tileIdx < ntiles;
  if (!active) tileIdx = ntiles - 1;       // clamp so barriers stay uniform
  const long row0 = (long)tileIdx * 16;

  const int wsel = (n < 2) ? n : 0;        // n>=2 columns compute unused garbage

  v8f c = {};

  for (int h0 = 0; h0 < H; h0 += TK) {
    // ---- stage 16 x TK hidden tile: fully coalesced float4 global loads ----
    const float* src = hs + row0 * (long)H + h0;
#pragma unroll
    for (int j = 0; j < 8; ++j) {
      const int r = 2 * j + half;          // lanes 0-15: row 2j, lanes 16-31: row 2j+1
      v4f v = *(const v4f*)(src + (long)r * H + n * 4);
      *(v4f*)&tile[w][r * TKP + n * 4] = v;
    }
    // ---- stage query / out_w chunks ----
    for (int t = lane; t < TK; t += 32) {
      wt[w][0][t] = q[h0 + t];
      wt[w][1][t] = ow[h0 + t];
    }
    __syncthreads();

    // ---- 16 WMMAs per chunk: K = 2*half + vgpr per ISA A/B fragment layout ----
#pragma unroll
    for (int kk = 0; kk < TK; kk += 4) {
      v2f a = *(const v2f*)&tile[w][n * TKP + kk + 2 * half];   // A[m=n][k]
      v2f b = *(const v2f*)&wt[w][wsel][kk + 2 * half];         // B[k][n]
      c = __builtin_amdgcn_wmma_f32_16x16x4_f32(
          /*neg_a=*/false, a, /*neg_b=*/false, b,
          /*c_mod=*/(short)0, c, /*reuse_a=*/false, /*reuse_b=*/false);
    }
    __syncthreads();
  }

  // D layout: lane holds D[m = r + 8*half][n] in c[r]. Columns 0/1 are the results.
  if (active && n < 2) {
    float* dst = (n == 0) ? sc : rv;
    const float mul = (n == 0) ? inv_scale : 1.0f;
#pragma unroll
    for (int r = 0; r < 8; ++r) {
      dst[row0 + r + 8 * half] = c[r] * mul;
    }
  }
}

// Pass 2 (tiny): per batch row, masked softmax over S and weighted sum of rv.
__global__ __launch_bounds__(256) void attnpool_softmax_out_kernel(
    const float* __restrict__ sc, const float* __restrict__ rv,
    const int* __restrict__ mask, const float* __restrict__ out_b,
    float* __restrict__ out, int S)
{
  const int b = blockIdx.x;
  const int t = (int)threadIdx.x;
  __shared__ float red[256];

  const float* scb = sc + (long)b * S;
  const float* rvb = rv + (long)b * S;
  const int*   mb  = mask + (long)b * S;

  float lm = -3.0e38f;
  for (int s = t; s < S; s += 256) {
    float v = (mb[s] != 0) ? scb[s] : -1.0e9f;
    lm = fmaxf(lm, v);
  }
  red[t] = lm; __syncthreads();
  for (int off = 128; off > 0; off >>= 1) {
    if (t < off) red[t] = fmaxf(red[t], red[t + off]);
    __syncthreads();
  }
  const float M = red[0]; __syncthreads();

  float le = 0.f, lr = 0.f;
  for (int s = t; s < S; s += 256) {
    float v = (mb[s] != 0) ? scb[s] : -1.0e9f;
    float e = __expf(v - M);
    le += e;
    lr += e * rvb[s];
  }
  red[t] = le; __syncthreads();
  for (int off = 128; off > 0; off >>= 1) {
    if (t < off) red[t] += red[t + off];
    __syncthreads();
  }
  const float Z = red[0]; __syncthreads();
  red[t] = lr; __syncthreads();
  for (int off = 128; off > 0; off >>= 1) {
    if (t < off) red[t] += red[t + off];
    __syncthreads();
  }
  if (t == 0) out[b] = red[0] / Z + out_b[0];
}

extern "C" void kernel_launch(void* const* d_in, const int* in_sizes, int n_in,
                              void* d_out, int out_size, void* d_ws, size_t ws_size,
                              hipStream_t stream) {
  const float* hs   = (const float*)d_in[0];  // [B,S,H]
  const int*   mask = (const int*)d_in[1];    // [B,S]
  const float* q    = (const float*)d_in[2];  // [H]
  const float* ow   = (const float*)d_in[3];  // [1,H]
  const float* ob   = (const float*)d_in[4];  // [1]
  float* out = (float*)d_out;                 // [B,1]

  const int H  = in_sizes[2];
  const int BS = in_sizes[1];
  const int B  = out_size;
  const int S  = BS / B;

  float* sc = (float*)d_ws;   // [BS]
  float* rv = sc + BS;        // [BS]

  const float inv_scale = 1.0f / sqrtf((float)H);

  const int ntiles = BS / 16;
  const int blocks = (ntiles + WAVES - 1) / WAVES;
  attnpool_scores_kernel<<<blocks, 256, 0, stream>>>(hs, q, ow, sc, rv, BS, H,
                                                     inv_scale);
  attnpool_softmax_out_kernel<<<B, 256, 0, stream>>>(sc, rv, mask, ob, out, S);
}